// GCN_LPA_5995774346009
// MI455X (gfx1250) — compile-verified
//
#include <hip/hip_runtime.h>

#define IN_C  128
#define HID_C 128
#define OUT_C 64
#define LPA_ITER 5

typedef __attribute__((ext_vector_type(2))) float v2f;
typedef __attribute__((ext_vector_type(8))) float v8f;

// ---------------- elementwise helpers ----------------

__global__ void zero_f32(float* __restrict__ p, long long n) {
  long long i = (long long)blockIdx.x * blockDim.x + threadIdx.x;
  if (i < n) p[i] = 0.0f;
}

__global__ void copy_f32(const float* __restrict__ s, float* __restrict__ d, long long n) {
  long long i = (long long)blockIdx.x * blockDim.x + threadIdx.x;
  if (i < n) d[i] = s[i];
}

__global__ void deg_kernel(const float* __restrict__ ea, const long long* __restrict__ ei,
                           int E, float* __restrict__ deg) {
  int e = blockIdx.x * blockDim.x + threadIdx.x;
  if (e < E) atomicAdd(&deg[ei[e]], ea[e]);
}

__global__ void val_kernel(const float* __restrict__ ea, const long long* __restrict__ ei,
                           int E, const float* __restrict__ deg, float* __restrict__ val) {
  int e = blockIdx.x * blockDim.x + threadIdx.x;
  if (e < E) {
    float d = deg[ei[e]];
    val[e] = (d == 0.0f) ? 0.0f : ea[e] / d;
  }
}

__global__ void bias_relu_kernel(const float* __restrict__ s, const float* __restrict__ b,
                                 float* __restrict__ out, long long total, int F) {
  long long i = (long long)blockIdx.x * blockDim.x + threadIdx.x;
  if (i < total) {
    float v = s[i] + b[(int)(i % F)];
    out[i] = v > 0.0f ? v : 0.0f;
  }
}

__global__ void bias_sigmoid_kernel(const float* __restrict__ s, const float* __restrict__ b,
                                    float* __restrict__ out, long long total, int F) {
  long long i = (long long)blockIdx.x * blockDim.x + threadIdx.x;
  if (i < total) {
    float v = s[i] + b[(int)(i % F)];
    out[i] = 1.0f / (1.0f + __expf(-v));
  }
}

__global__ void sigmoid_kernel(const float* __restrict__ s, float* __restrict__ out,
                               long long total) {
  long long i = (long long)blockIdx.x * blockDim.x + threadIdx.x;
  if (i < total) out[i] = 1.0f / (1.0f + __expf(-s[i]));
}

// ---------------- dense GEMM via V_WMMA_F32_16X16X4_F32 ----------------
// One wave computes one 16x16 tile of C = A[MxK] * B[KxNW].
// f32 A-fragment layout (ISA 7.12.2): lane m = lane&15; VGPR0 holds
// K = 2*(lane>>4), VGPR1 holds K = 2*(lane>>4)+1. B mirrored (n = lane&15).
// C/D layout: VGPR r holds row = tile_m*16 + r + 8*(lane>>4), col = lane&15.

template<int K, int NW>
__global__ void gemm_wmma(const float* __restrict__ A, const float* __restrict__ B,
                          float* __restrict__ C, int M) {
  const int ntiles = NW / 16;
  int wave = blockIdx.x * (blockDim.x >> 5) + (threadIdx.x >> 5);
  int totalTiles = (M >> 4) * ntiles;
  if (wave >= totalTiles) return;          // wave-uniform: EXEC stays all-1s
  int lane  = threadIdx.x & 31;
  int tm    = wave / ntiles;
  int tn    = wave % ntiles;
  int mrow  = (tm << 4) + (lane & 15);
  int ncol  = (tn << 4) + (lane & 15);
  int khalf = (lane >> 4) << 1;            // 0 for lanes 0-15, 2 for lanes 16-31

  const float* arow = A + (long long)mrow * K;
  const float* bcol = B + ncol;

  v8f acc = {};
#pragma unroll 4
  for (int kk = 0; kk < K; kk += 4) {
    v2f a, b;
    a.x = arow[kk + khalf];
    a.y = arow[kk + khalf + 1];
    b.x = bcol[(long long)(kk + khalf) * NW];
    b.y = bcol[(long long)(kk + khalf + 1) * NW];
    acc = __builtin_amdgcn_wmma_f32_16x16x4_f32(
        /*neg_a=*/false, a, /*neg_b=*/false, b,
        /*c_mod=*/(short)0, acc, /*reuse_a=*/false, /*reuse_b=*/false);
  }

  long long base = ((long long)(tm << 4) + ((lane >> 4) << 3)) * NW + ncol;
#pragma unroll
  for (int r = 0; r < 8; ++r)
    C[base + (long long)r * NW] = acc[r];
}

// ---------------- SpMM: out[row] += val[e] * in[col], edge-parallel ----------------
// Each lane handles 4 consecutive features (b128 gather + 4 clustered f32 atomics).
// blockDim = (F/4, 256/(F/4)); a y-group's lanes cover one contiguous row segment.

template<int F>
__global__ void spmm_kernel(const float* __restrict__ val, const long long* __restrict__ ei,
                            int E, const float* __restrict__ vin, float* __restrict__ vout) {
  int e = blockIdx.x * blockDim.y + threadIdx.y;
  if (e >= E) return;
  long long r = ei[e];
  long long c = ei[(long long)E + e];
  float v = val[e];
  int f = threadIdx.x << 2;
  const float4 s = *(const float4*)(vin + c * F + f);
  float* dst = vout + r * F + f;
  atomicAdd(dst + 0, v * s.x);
  atomicAdd(dst + 1, v * s.y);
  atomicAdd(dst + 2, v * s.z);
  atomicAdd(dst + 3, v * s.w);
}

// ---------------- driver ----------------

static inline int cdiv_ll(long long a, long long b) { return (int)((a + b - 1) / b); }

extern "C" void kernel_launch(void* const* d_in, const int* in_sizes, int n_in,
                              void* d_out, int out_size, void* d_ws, size_t ws_size,
                              hipStream_t stream) {
  const float*     x    = (const float*)d_in[0];
  const float*     soft = (const float*)d_in[1];
  const float*     ea   = (const float*)d_in[2];
  const float*     w1   = (const float*)d_in[3];
  const float*     b1   = (const float*)d_in[4];
  const float*     w2   = (const float*)d_in[5];
  const float*     b2   = (const float*)d_in[6];
  const long long* ei   = (const long long*)d_in[7];

  const int N = in_sizes[0] / IN_C;
  const int E = in_sizes[2];
  float* out = (float*)d_out;

  // workspace layout
  float* deg  = (float*)d_ws;                      // N
  float* val  = deg + N;                           // E
  float* bufA = val + E;                           // N * HID_C
  float* bufB = bufA + (long long)N * HID_C;       // N * HID_C

  const long long nh = (long long)N * HID_C;       // N*128
  const long long no = (long long)N * OUT_C;       // N*64

  // 1. degree + edge normalization
  zero_f32<<<cdiv_ll(N, 256), 256, 0, stream>>>(deg, N);
  deg_kernel<<<cdiv_ll(E, 256), 256, 0, stream>>>(ea, ei, E, deg);
  val_kernel<<<cdiv_ll(E, 256), 256, 0, stream>>>(ea, ei, E, deg, val);

  // 2. xw1 = x @ w1  (WMMA)
  {
    int tiles = (N / 16) * (HID_C / 16);
    gemm_wmma<IN_C, HID_C><<<cdiv_ll(tiles, 8), 256, 0, stream>>>(x, w1, bufA, N);
  }

  // 3. s1 = A_n @ xw1 ; h = relu(s1 + b1)  (h overwrites bufA)
  zero_f32<<<cdiv_ll(nh, 256), 256, 0, stream>>>(bufB, nh);
  spmm_kernel<HID_C><<<cdiv_ll(E, 8), dim3(HID_C / 4, 8), 0, stream>>>(val, ei, E, bufA, bufB);
  bias_relu_kernel<<<cdiv_ll(nh, 256), 256, 0, stream>>>(bufB, b1, bufA, nh, HID_C);

  // 4. hw2 = h @ w2  (WMMA) -> bufB[0 : N*64)
  {
    int tiles = (N / 16) * (OUT_C / 16);
    gemm_wmma<HID_C, OUT_C><<<cdiv_ll(tiles, 8), 256, 0, stream>>>(bufA, w2, bufB, N);
  }

  // 5. s2 = A_n @ hw2 ; x_out = sigmoid(s2 + b2)
  float* s2 = bufB + no;
  zero_f32<<<cdiv_ll(no, 256), 256, 0, stream>>>(s2, no);
  spmm_kernel<OUT_C><<<cdiv_ll(E, 16), dim3(OUT_C / 4, 16), 0, stream>>>(val, ei, E, bufB, s2);
  bias_sigmoid_kernel<<<cdiv_ll(no, 256), 256, 0, stream>>>(s2, b2, out, no, OUT_C);

  // 6. label propagation: 5x labels = A_n @ labels, then sigmoid
  float* ping = bufA;            // bufA no longer needed: reuse as 2 x N*64
  float* pong = bufA + no;
  copy_f32<<<cdiv_ll(no, 256), 256, 0, stream>>>(soft, ping, no);
  for (int it = 0; it < LPA_ITER; ++it) {
    zero_f32<<<cdiv_ll(no, 256), 256, 0, stream>>>(pong, no);
    spmm_kernel<OUT_C><<<cdiv_ll(E, 16), dim3(OUT_C / 4, 16), 0, stream>>>(val, ei, E, ping, pong);
    float* t = ping; ping = pong; pong = t;
  }
  sigmoid_kernel<<<cdiv_ll(no, 256), 256, 0, stream>>>(ping, out + no, no);
}